// LPRadon_66898410603172
// MI455X (gfx1250) — compile-verified
//
#include <hip/hip_runtime.h>
#include <math.h>
#include <stdint.h>

// ---------------------------------------------------------------------------
// LPRadon on MI455X (gfx1250): log-polar gather -> 2D DFT-convolution done as
// dense DFT GEMMs on V_WMMA_F32_16X16X4_F32 with async global->LDS staging
// (GLOBAL_LOAD_ASYNC_TO_LDS_B128) -> scatter to sinogram.
// ---------------------------------------------------------------------------

typedef __attribute__((ext_vector_type(2))) float v2f;
typedef __attribute__((ext_vector_type(8))) float v8f;

#define NPIX   512
#define NRHO   1024
#define NTH    512
#define NBATCH 16
#define PLANE  (NRHO * NTH)            // 524288
#define TOTAL  (NBATCH * PLANE)        // 8388608
#define PIF    3.14159265358979323846f

static __device__ __forceinline__ int clampi(int v, int lo, int hi) {
    return v < lo ? lo : (v > hi ? hi : v);
}

// Async copy 16B from global memory directly into LDS (no VGPR bounce).
// ldsOff is the wave-relative LDS byte offset (flat LDS aperture: addr[31:0]).
static __device__ __forceinline__ void async_g2l_b128(uint32_t ldsOff, const float* gaddr) {
    asm volatile("global_load_async_to_lds_b128 %0, %1, off"
                 :: "v"(ldsOff), "v"(gaddr)
                 : "memory");
}
static __device__ __forceinline__ void wait_async0() {
    asm volatile("s_wait_asynccnt 0x0" ::: "memory");
}

// -------------------------- constant generation ----------------------------

__constant__ float kCorrNum[11] = {
    -216254335.f, 679543284.f, -1412947389.f, 2415881496.f, -3103579086.f,
    2939942400.f, -2023224114.f, 984515304.f, -321455811.f, 63253516.f, -5675265.f};

__device__ __forceinline__ float hval(int j) {
    const float D = 958003200.f;
    if (j == 0) return 1.f + 2.f * kCorrNum[0] / D;   // 2*(c0-0.5)
    if (j <= 10) return 1.f + kCorrNum[j] / D;
    int jr = 2048 - j;
    if (jr >= 1 && jr <= 10) return 1.f + kCorrNum[jr] / D;
    return 1.f;
}

// DFT matrices Cm[i*n+j]=cos(2*pi*i*j/n), Sm=sin(...). Exact arg reduction via
// integer modulus so the angle stays in [0, 2*pi).
__global__ void init_twiddle(float* __restrict__ Cm, float* __restrict__ Sm, int n) {
    int idx = blockIdx.x * blockDim.x + threadIdx.x;
    if (idx >= n * n) return;
    int i = idx / n, j = idx - i * n;
    int m = (int)(((long long)i * (long long)j) % (long long)n);
    float ang = 6.28318530717958647692f * (float)m / (float)n;
    float s, c;
    __sincosf(ang, &s, &c);
    Cm[idx] = c;
    Sm[idx] = s;
}

// zeta[fr][ft], fr in 0..1023, ft in 0..256 (rfft half-spectrum), interleaved
// re/im. fcosa = cos(th)^expo = (1/cos th)*e^{-i*2pi*kr*ln(cos th)/P}; direct
// 2048-term row DFT with both fftshifts folded in ((-1)^ft factor, signed kr),
// divided by the closed-form B3-spline spectra (2+cos)/3.
__global__ void init_zeta(float* __restrict__ zt, float beta, float P, float constZ) {
    int idx = blockIdx.x * blockDim.x + threadIdx.x;
    if (idx >= 1024 * 257) return;
    int fr = idx / 257, ft = idx - fr * 257;
    float re = 0.f, im = 0.f;
    if (fr != 512 && ft != 256) {
        int kr = (fr <= 511) ? fr : fr - 1024;
        float wk = 6.28318530717958f * (float)kr / P;
        float wf = PIF * (float)ft / 1024.f;              // 2*pi*ft/2048
        float dth_l = beta / 1024.f;
        for (int j = 0; j < 2048; ++j) {
            float th = (float)(j - 1024) * dth_l;         // |th| <= beta, cos>0
            float c = __cosf(th);
            float L = __logf(c);
            float ph = wk * L + wf * (float)j;
            float s, co;
            __sincosf(ph, &s, &co);
            float amp = hval(j) / c;
            re += amp * co;
            im -= amp * s;
        }
        float sign = (ft & 1) ? -1.f : 1.f;
        float b3r = (2.f + __cosf(6.283185307f * (float)fr / 1024.f)) / 3.f;
        float b3t = (2.f + __cosf(6.283185307f * (float)ft / 512.f)) / 3.f;
        float scale = sign * dth_l * constZ / (b3r * b3t);
        re *= scale;
        im *= scale;
    }
    zt[2 * idx] = re;
    zt[2 * idx + 1] = im;
}

// Expand half-spectrum zeta to the full Hermitian 1024x512 multiplier.
__global__ void expand_zeta(const float* __restrict__ zt,
                            float* __restrict__ Zre, float* __restrict__ Zim) {
    int idx = blockIdx.x * blockDim.x + threadIdx.x;
    if (idx >= 1024 * 512) return;
    int fr = idx >> 9, ft = idx & 511;
    float re, im;
    if (ft <= 256) {
        re = zt[2 * (fr * 257 + ft)];
        im = zt[2 * (fr * 257 + ft) + 1];
    } else {
        int fr2 = (1024 - fr) & 1023;
        int ft2 = 512 - ft;
        re = zt[2 * (fr2 * 257 + ft2)];
        im = -zt[2 * (fr2 * 257 + ft2) + 1];
    }
    Zre[idx] = re;
    Zim[idx] = im;
}

__global__ void zero_f32(float* __restrict__ p, int n) {
    int i = blockIdx.x * blockDim.x + threadIdx.x;
    if (i < n) p[i] = 0.f;
}

// ------------------------------ gather stage -------------------------------

__global__ void gather_lp(const float* __restrict__ x, float* __restrict__ lp,
                          float drho, float dth, float aR, float cA, float sA) {
    int idx = blockIdx.x * blockDim.x + threadIdx.x;
    if (idx >= TOTAL) return;
    int b = idx >> 19;
    int rt = idx & (PLANE - 1);
    int r = rt >> 9, t = rt & 511;
    float er = __expf((float)(r - 1024) * drho);
    float th = (float)(t - 256) * dth;
    float s, c;
    __sincosf(th, &s, &c);
    float t1 = er * c - (1.f - aR);
    float t2 = er * s;
    float l1 = (t1 * cA - t2 * sA) / aR;
    float l2 = -((t1 * sA + t2 * cA) / aR);
    l1 = (l1 + 1.f) * 0.5f * 511.f;
    l2 = (l2 + 1.f) * 0.5f * 511.f;
    int y = clampi((int)floorf(l1), 0, 510);
    int xx = clampi((int)floorf(l2), 0, 510);
    lp[idx] = x[b * (NPIX * NPIX) + y * NPIX + xx] * er;
}

// ------------------------- WMMA dual-GEMM kernel ---------------------------
// C = s1*A1*B1 + s2*A2*B2  (row-major; M%64==0, N%128==0, K%32==0).
// Block = 128 threads (4 waves), workgroup tile 64x128, each wave owns a
// 32x64 strip as 2x4 V_WMMA_F32_16X16X4_F32 accumulators (4x A-frag reuse).
// Tiles staged with GLOBAL_LOAD_ASYNC_TO_LDS_B128 (ASYNCcnt), scales applied
// to A fragments in-register. LDS rows padded to 16B-aligned strides that are
// bank-conflict-free for the strided A-column reads.
__global__ __launch_bounds__(128) void cgemm2_wmma(
    const float* __restrict__ A1, const float* __restrict__ B1, float s1,
    const float* __restrict__ A2, const float* __restrict__ B2, float s2,
    float* __restrict__ Cmat, int M, int Nn, int K,
    long aBatch, long bBatch, long cBatch, int dual) {
    __shared__ float As1[64][36];   // 36-float stride: 16B aligned, conflict-free
    __shared__ float As2[64][36];
    __shared__ float Bs1[32][132];  // 132-float stride: 16B aligned
    __shared__ float Bs2[32][132];

    const int bz = blockIdx.z;
    const float* A1p = A1 + (long)bz * aBatch;
    const float* B1p = B1 + (long)bz * bBatch;
    const float* A2p = A2 + (long)bz * aBatch;
    const float* B2p = B2 + (long)bz * bBatch;
    float* Cp = Cmat + (long)bz * cBatch;

    const uint32_t ldsA1 = (uint32_t)(uintptr_t)&As1[0][0];
    const uint32_t ldsA2 = (uint32_t)(uintptr_t)&As2[0][0];
    const uint32_t ldsB1 = (uint32_t)(uintptr_t)&Bs1[0][0];
    const uint32_t ldsB2 = (uint32_t)(uintptr_t)&Bs2[0][0];

    const int tileM = blockIdx.y * 64;
    const int tileN = blockIdx.x * 128;
    const int tid = threadIdx.x;
    const int lane = tid & 31;
    const int wave = tid >> 5;
    const int wr = wave >> 1;  // M strip (0..1)
    const int wc = wave & 1;   // N strip (0..1)

    v8f acc[2][4] = {};

    const int lm = lane & 15;   // m (A) / n (B) index within a 16-wide tile
    const int lpr = lane >> 4;  // K-pair selector

    for (int k0 = 0; k0 < K; k0 += 32) {
        // ---- async stage A tiles: 64x32 floats = 512 16B-chunks, 4/thread
#pragma unroll
        for (int i = 0; i < 4; ++i) {
            int c = i * 128 + tid;
            int row = c >> 3, cc = (c & 7) * 4;
            async_g2l_b128(ldsA1 + (uint32_t)(row * 36 + cc) * 4u,
                           A1p + (long)(tileM + row) * K + k0 + cc);
            if (dual)
                async_g2l_b128(ldsA2 + (uint32_t)(row * 36 + cc) * 4u,
                               A2p + (long)(tileM + row) * K + k0 + cc);
        }
        // ---- async stage B tiles: 32x128 floats = 1024 16B-chunks, 8/thread
#pragma unroll
        for (int i = 0; i < 8; ++i) {
            int c = i * 128 + tid;
            int row = c >> 5, cc = (c & 31) * 4;
            async_g2l_b128(ldsB1 + (uint32_t)(row * 132 + cc) * 4u,
                           B1p + (long)(k0 + row) * Nn + tileN + cc);
            if (dual)
                async_g2l_b128(ldsB2 + (uint32_t)(row * 132 + cc) * 4u,
                               B2p + (long)(k0 + row) * Nn + tileN + cc);
        }
        wait_async0();
        __syncthreads();

        const int m0 = wr * 32;
        const int n0 = wc * 64;
#pragma unroll
        for (int kk = 0; kk < 32; kk += 4) {
            // A frag 16x4: lanes 0-15 hold K={0,1}, lanes 16-31 hold K={2,3}.
            // B frag 4x16: same K-pair split, N striped across 16 lanes.
            v2f af[2], bf[4];
#pragma unroll
            for (int ri = 0; ri < 2; ++ri) {
                af[ri].x = As1[m0 + ri * 16 + lm][kk + 2 * lpr];
                af[ri].y = As1[m0 + ri * 16 + lm][kk + 2 * lpr + 1];
                af[ri] *= s1;
            }
#pragma unroll
            for (int ci = 0; ci < 4; ++ci) {
                bf[ci].x = Bs1[kk + 2 * lpr][n0 + ci * 16 + lm];
                bf[ci].y = Bs1[kk + 2 * lpr + 1][n0 + ci * 16 + lm];
            }
#pragma unroll
            for (int ri = 0; ri < 2; ++ri)
#pragma unroll
                for (int ci = 0; ci < 4; ++ci)
                    acc[ri][ci] = __builtin_amdgcn_wmma_f32_16x16x4_f32(
                        false, af[ri], false, bf[ci], (short)0, acc[ri][ci],
                        false, false);
            if (dual) {
                v2f af2[2], bf2[4];
#pragma unroll
                for (int ri = 0; ri < 2; ++ri) {
                    af2[ri].x = As2[m0 + ri * 16 + lm][kk + 2 * lpr];
                    af2[ri].y = As2[m0 + ri * 16 + lm][kk + 2 * lpr + 1];
                    af2[ri] *= s2;
                }
#pragma unroll
                for (int ci = 0; ci < 4; ++ci) {
                    bf2[ci].x = Bs2[kk + 2 * lpr][n0 + ci * 16 + lm];
                    bf2[ci].y = Bs2[kk + 2 * lpr + 1][n0 + ci * 16 + lm];
                }
#pragma unroll
                for (int ri = 0; ri < 2; ++ri)
#pragma unroll
                    for (int ci = 0; ci < 4; ++ci)
                        acc[ri][ci] = __builtin_amdgcn_wmma_f32_16x16x4_f32(
                            false, af2[ri], false, bf2[ci], (short)0, acc[ri][ci],
                            false, false);
            }
        }
        __syncthreads();
    }

    // C/D layout: VGPR v holds row (v + 8*(lane>=16)), N = lane&15.
    const int lh = lane >> 4;
#pragma unroll
    for (int ri = 0; ri < 2; ++ri)
#pragma unroll
        for (int ci = 0; ci < 4; ++ci)
#pragma unroll
            for (int v = 0; v < 8; ++v) {
                int row = tileM + wr * 32 + ri * 16 + v + 8 * lh;
                int col = tileN + wc * 64 + ci * 16 + lm;
                Cp[(long)row * Nn + col] = acc[ri][ci][v];
            }
}

// ------------------------- pointwise zeta multiply -------------------------

__global__ void zmul(float* __restrict__ Fre, float* __restrict__ Fim,
                     const float* __restrict__ Zre, const float* __restrict__ Zim) {
    int idx = blockIdx.x * blockDim.x + threadIdx.x;
    if (idx >= TOTAL) return;
    int zi = idx & (PLANE - 1);
    float a = Fre[idx], b = Fim[idx];
    float zr = Zre[zi], zim = Zim[zi];
    Fre[idx] = a * zr - b * zim;
    Fim[idx] = a * zim + b * zr;
}

// ------------------------------ sample stage -------------------------------

__global__ void sample_out(const float* __restrict__ gk, float* __restrict__ out,
                           int k, float beta, float dth, float drho, float aR) {
    int idx = blockIdx.x * blockDim.x + threadIdx.x;
    if (idx >= TOTAL) return;
    int b = idx >> 19;
    int p = idx & (PLANE - 1);
    int i = p >> 9, j = p & 511;
    float th0 = (float)i * (PIF / 1024.f) - beta * 0.5f;
    float lo = (float)k * beta - beta * 0.5f;
    float hi = (float)k * beta + beta * 0.5f;
    if (th0 < lo || th0 >= hi) return;
    float th00 = th0 - (float)k * beta;
    float s0 = -1.f + 2.f * (float)j / 511.f;
    float p1 = th00 / dth + 256.f;
    float v = s0 * aR + (1.f - aR) * __cosf(th00);
    float p2r = (v > 0.f) ? __logf(v) : 0.f;
    float p2 = p2r / drho + 1024.f;
    int y = clampi((int)floorf(p2), 0, 1022);
    int xx = clampi((int)floorf(p1), 0, 510);
    out[idx] = gk[b * PLANE + y * NTH + xx];
}

// ------------------------------- host driver -------------------------------

extern "C" void kernel_launch(void* const* d_in, const int* in_sizes, int n_in,
                              void* d_out, int out_size, void* d_ws, size_t ws_size,
                              hipStream_t stream) {
    (void)in_sizes; (void)n_in; (void)out_size; (void)ws_size;
    const float* x = (const float*)d_in[0];
    float* out = (float*)d_out;
    float* ws = (float*)d_ws;

    // Host-side scalar precompute (deterministic, graph-safe).
    const double PId = 3.14159265358979323846;
    const double beta = PId / 3.0;
    const double aR = sin(beta / 2) / (1 + sin(beta / 2));        // 1/3
    const double am = (cos(beta / 2) - sin(beta / 2)) / (1 + sin(beta / 2));
    double g = -1e300;
    for (int i = 0; i < 1000; ++i) {
        double t = -PId / 2 + PId * (double)i / 999.0;
        double wre = aR * cos(t) + (1 - aR), wim = aR * sin(t);
        double v = log(sqrt(wre * wre + wim * wim)) +
                   log(cos(beta / 2 - atan2(wim, wre)));
        if (v == v && v > g) g = v;
    }
    const double P = g - log(am);
    const float dth = (float)(2.0 * beta / NTH);
    const float drho = (float)(P / NRHO);
    const float constZ = (float)(sqrt((double)NPIX / 1024.0) * PId / 4.0 / aR / sqrt(2.0));

    // Workspace layout (floats).
    size_t oC512 = 0;
    size_t oS512 = oC512 + 512 * 512;
    size_t oC1024 = oS512 + 512 * 512;
    size_t oS1024 = oC1024 + 1024 * 1024;
    size_t oZre = oS1024 + 1024 * 1024;
    size_t oZim = oZre + (size_t)PLANE;
    size_t oZt = oZim + (size_t)PLANE;
    size_t oLP = oZt + 1024 * 257 * 2;
    size_t oT0 = oLP + (size_t)TOTAL;  // FthRe / GthRe
    size_t oT1 = oT0 + (size_t)TOTAL;  // FthIm / GthIm
    size_t oT2 = oT1 + (size_t)TOTAL;  // FRe
    size_t oT3 = oT2 + (size_t)TOTAL;  // FIm

    const int TB = 256;
    init_twiddle<<<(512 * 512 + TB - 1) / TB, TB, 0, stream>>>(ws + oC512, ws + oS512, 512);
    init_twiddle<<<(1024 * 1024 + TB - 1) / TB, TB, 0, stream>>>(ws + oC1024, ws + oS1024, 1024);
    init_zeta<<<(1024 * 257 + TB - 1) / TB, TB, 0, stream>>>(ws + oZt, (float)beta, (float)P, constZ);
    expand_zeta<<<(PLANE + TB - 1) / TB, TB, 0, stream>>>(ws + oZt, ws + oZre, ws + oZim);
    zero_f32<<<(TOTAL + TB - 1) / TB, TB, 0, stream>>>(out, TOTAL);

    const dim3 gemmBlk(128);
    const dim3 gemmGrid(512 / 128, 1024 / 64, NBATCH);  // N tiles, M tiles, batch
    const float invScale = 1.0f / (float)(NRHO * NTH);

    for (int k = 0; k < 3; ++k) {
        const float a = (float)((double)k * beta + beta / 2.0);
        const float cA = cosf(a), sA = sinf(a);

        gather_lp<<<(TOTAL + TB - 1) / TB, TB, 0, stream>>>(
            x, ws + oLP, drho, dth, (float)aR, cA, sA);

        // Row rfft (full complex): Fth = lp * (C512 - i*S512)
        cgemm2_wmma<<<gemmGrid, gemmBlk, 0, stream>>>(
            ws + oLP, ws + oC512, 1.f, ws + oLP, ws + oC512, 0.f,
            ws + oT0, 1024, 512, 512, (long)PLANE, 0L, (long)PLANE, 0);
        cgemm2_wmma<<<gemmGrid, gemmBlk, 0, stream>>>(
            ws + oLP, ws + oS512, -1.f, ws + oLP, ws + oS512, 0.f,
            ws + oT1, 1024, 512, 512, (long)PLANE, 0L, (long)PLANE, 0);

        // Column DFT: F = (C1024 - i*S1024) x Fth
        cgemm2_wmma<<<gemmGrid, gemmBlk, 0, stream>>>(
            ws + oC1024, ws + oT0, 1.f, ws + oS1024, ws + oT1, 1.f,
            ws + oT2, 1024, 512, 1024, 0L, (long)PLANE, (long)PLANE, 1);
        cgemm2_wmma<<<gemmGrid, gemmBlk, 0, stream>>>(
            ws + oC1024, ws + oT1, 1.f, ws + oS1024, ws + oT0, -1.f,
            ws + oT3, 1024, 512, 1024, 0L, (long)PLANE, (long)PLANE, 1);

        // Pointwise zeta.
        zmul<<<(TOTAL + TB - 1) / TB, TB, 0, stream>>>(
            ws + oT2, ws + oT3, ws + oZre, ws + oZim);

        // Inverse column DFT: Gth = (C1024 + i*S1024) x F
        cgemm2_wmma<<<gemmGrid, gemmBlk, 0, stream>>>(
            ws + oC1024, ws + oT2, 1.f, ws + oS1024, ws + oT3, -1.f,
            ws + oT0, 1024, 512, 1024, 0L, (long)PLANE, (long)PLANE, 1);
        cgemm2_wmma<<<gemmGrid, gemmBlk, 0, stream>>>(
            ws + oC1024, ws + oT3, 1.f, ws + oS1024, ws + oT2, 1.f,
            ws + oT1, 1024, 512, 1024, 0L, (long)PLANE, (long)PLANE, 1);

        // Inverse row DFT (real part), scaled: gk = (GthRe*C512 - GthIm*S512)/(1024*512)
        cgemm2_wmma<<<gemmGrid, gemmBlk, 0, stream>>>(
            ws + oT0, ws + oC512, invScale, ws + oT1, ws + oS512, -invScale,
            ws + oLP, 1024, 512, 512, (long)PLANE, 0L, (long)PLANE, 1);

        sample_out<<<(TOTAL + TB - 1) / TB, TB, 0, stream>>>(
            ws + oLP, out, k, (float)beta, dth, drho, (float)aR);
    }
}